// PositionCosine_6828998001014
// MI455X (gfx1250) — compile-verified
//
#include <hip/hip_runtime.h>

typedef __attribute__((ext_vector_type(2))) float v2f;
typedef __attribute__((ext_vector_type(8))) float v8f;

#define B_    16
#define C_    64
#define S_    512
#define D_    256
#define UNK_  1
#define SB_   8      // s-partitions across blocks
#define WAVES_ 8     // waves per block (256 threads, wave32)

// scores[b,c] partials via V_WMMA_F32_16X16X4_F32.
// A (16x4) = query-embedding chunk broadcast over M; B (4x16) = 16 gathered
// source-embedding rows (one per c in the tile). D row 0 = 16 partial scores.
__global__ __launch_bounds__(256) void
score_kernel(const int* __restrict__ sources, const int* __restrict__ queries,
             const float* __restrict__ emb, int vocab,
             float* __restrict__ partial)
{
    const int b    = blockIdx.x;
    const int ct   = blockIdx.y;
    const int sb   = blockIdx.z;
    const int tid  = threadIdx.x;
    const int wave = tid >> 5;
    const int lane = tid & 31;
    const int cl   = lane & 15;           // c within the 16-wide tile
    const int h    = (lane >> 4) & 1;     // K-half selector per ISA layout
    const int c    = ct * 16 + cl;
    const long src_row = (long)(b * C_ + c) * S_;

    const int ns = S_ / SB_ / WAVES_;     // 8 s-values per wave
    const int s0 = sb * (S_ / SB_) + wave * ns;

    v8f acc = {};
    for (int si = 0; si < ns; ++si) {
        const int s = s0 + si;
        int q = queries[b * S_ + s];      // wave-uniform -> uniform branch
        if (q == 0) continue;             // qmask: skip zero-suffix positions
        if ((unsigned)q >= (unsigned)vocab) q = UNK_;
        int sid = sources[src_row + s];
        if ((unsigned)sid >= (unsigned)vocab) sid = UNK_;

        const float* qrow = emb + (long)q   * D_;
        const float* srow = emb + (long)sid * D_;

#pragma unroll 8
        for (int d0 = 0; d0 < D_; d0 += 4) {
            // A: lanes<16 get eqm[d0,d0+1], lanes>=16 get eqm[d0+2,d0+3]
            v2f a  = *(const v2f*)(qrow + d0 + 2 * h);
            // B: lane n / n+16 hold the two b64 halves of row src[c_n]
            v2f bm = *(const v2f*)(srow + d0 + 2 * h);
            acc = __builtin_amdgcn_wmma_f32_16x16x4_f32(
                false, a, false, bm, (short)0, acc, false, false);
        }
    }

    // D[M=0, N=n] lives in acc[0] on lanes 0..15 -> block reduce over waves
    __shared__ float red[WAVES_ * 16];
    if (lane < 16) red[wave * 16 + cl] = acc[0];
    __syncthreads();
    if (tid < 16) {
        float sum = 0.f;
        for (int w = 0; w < WAVES_; ++w) sum += red[w * 16 + tid];
        partial[(long)sb * (B_ * C_) + b * C_ + ct * 16 + tid] = sum;
    }
}

// Sum partials, softmax over contexts, argmax, copy winning source row.
// d_out layout: [0, B*S) = out_sources (as float), [B*S, B*S + B*C) = sims.
__global__ __launch_bounds__(64) void
finalize_kernel(const int* __restrict__ sources,
                const float* __restrict__ partial,
                float* __restrict__ out)
{
    const int b = blockIdx.x;
    const int t = threadIdx.x;            // 64 threads = 64 contexts

    float sc = 0.f;
    for (int p = 0; p < SB_; ++p) sc += partial[p * (B_ * C_) + b * C_ + t];

    __shared__ float sh[C_];
    __shared__ float smax, ssum;
    __shared__ int   stop;

    sh[t] = sc;
    __syncthreads();
    if (t == 0) {
        float m = sh[0]; int mi = 0;
        for (int i = 1; i < C_; ++i)
            if (sh[i] > m) { m = sh[i]; mi = i; }   // first-max, like argmax
        smax = m; stop = mi;
    }
    __syncthreads();
    float e = expf(sc - smax);
    sh[t] = e;
    __syncthreads();
    if (t == 0) {
        float s2 = 0.f;
        for (int i = 0; i < C_; ++i) s2 += sh[i];
        ssum = s2;
    }
    __syncthreads();

    out[B_ * S_ + b * C_ + t] = e / ssum;           // sims

    const int top = stop;
    const long row = (long)(b * C_ + top) * S_;
    for (int i = t; i < S_; i += C_)
        out[b * S_ + i] = (float)sources[row + i];  // out_sources
}

extern "C" void kernel_launch(void* const* d_in, const int* in_sizes, int n_in,
                              void* d_out, int out_size, void* d_ws, size_t ws_size,
                              hipStream_t stream)
{
    const int*   sources = (const int*)d_in[0];
    const int*   queries = (const int*)d_in[1];
    // d_in[2] = context_len scalar (== C_, compile-time constant here)
    const float* emb     = (const float*)d_in[3];
    const int vocab = in_sizes[3] / D_;

    float* partial = (float*)d_ws;   // SB_ * B_ * C_ floats = 32 KB

    dim3 grid(B_, C_ / 16, SB_);
    score_kernel<<<grid, 256, 0, stream>>>(sources, queries, emb, vocab, partial);
    finalize_kernel<<<B_, C_, 0, stream>>>(sources, partial, (float*)d_out);
}